// GCNNet1_45749991637434
// MI455X (gfx1250) — compile-verified
//
#include <hip/hip_runtime.h>
#include <hip/hip_bf16.h>

typedef __attribute__((ext_vector_type(2))) float v2f;
typedef __attribute__((ext_vector_type(8))) float v8f;

#define N_NODES 100000

// ---------------------------------------------------------------------------
// Degree / normalization
// ---------------------------------------------------------------------------
__global__ void deg_init(float* deg, int n) {
    int i = blockIdx.x * blockDim.x + threadIdx.x;
    if (i < n) deg[i] = 1.0f;                       // self-loop contribution
}

__global__ void deg_accum(const long long* __restrict__ dst, float* deg, long long E) {
    long long e = (long long)blockIdx.x * blockDim.x + threadIdx.x;
    if (e < E) atomicAdd(&deg[dst[e]], 1.0f);
}

__global__ void deg_to_dinv(float* deg, int n) {
    int i = blockIdx.x * blockDim.x + threadIdx.x;
    if (i < n) {
        float d = deg[i];
        deg[i] = (d > 0.0f) ? rsqrtf(d) : 0.0f;
    }
}

// ---------------------------------------------------------------------------
// Dense transform: H = (relu?)A @ W   using V_WMMA_F32_16X16X4_F32
// One wave (32 threads) computes one 16x16 output tile.
//   A operand (16x4 f32): lane<16 -> M=lane, K={k0,k0+1}; lane>=16 -> M=lane-16, K={k0+2,k0+3}
//   B operand (4x16 f32): lane<16 -> N=lane, K={k0,k0+1}; lane>=16 -> N=lane-16, K={k0+2,k0+3}
//   D (16x16 f32): vgpr r -> M = r + 8*(lane>=16), N = lane&15
// ---------------------------------------------------------------------------
__global__ void gemm_wmma_f32(const float* __restrict__ A, const float* __restrict__ W,
                              float* __restrict__ H, int K, int Nf, int relu_in) {
    const int lane = threadIdx.x;          // 0..31, exactly one wave32, EXEC all ones
    const int half = lane >> 4;            // 0 or 1
    const int l16  = lane & 15;

    const int mrow = blockIdx.x * 16 + l16;            // A row for this lane
    const int ncol = blockIdx.y * 16 + l16;            // B/D column for this lane
    const int ncs  = (ncol < Nf) ? ncol : (Nf - 1);    // clamped (avoid OOB), zero-masked below
    const float nmask = (ncol < Nf) ? 1.0f : 0.0f;

    v8f acc = {};
    for (int k0 = 0; k0 < K; k0 += 4) {
        const int ka = k0 + half * 2;      // this lane's K base (Fin always multiple of 4)

        float a0 = A[(size_t)mrow * K + ka];
        float a1 = A[(size_t)mrow * K + ka + 1];
        if (relu_in) { a0 = fmaxf(a0, 0.0f); a1 = fmaxf(a1, 0.0f); }
        v2f av; av.x = a0; av.y = a1;

        float b0 = W[(size_t)ka * Nf + ncs] * nmask;
        float b1 = W[(size_t)(ka + 1) * Nf + ncs] * nmask;
        v2f bv; bv.x = b0; bv.y = b1;

        // (neg_a, A, neg_b, B, c_mod, C, reuse_a, reuse_b)
        acc = __builtin_amdgcn_wmma_f32_16x16x4_f32(false, av, false, bv,
                                                    (short)0, acc, false, false);
    }

    if (ncol < Nf) {
#pragma unroll
        for (int r = 0; r < 8; ++r) {
            int row = blockIdx.x * 16 + r + half * 8;
            H[(size_t)row * Nf + ncol] = acc[r];
        }
    }
}

// ---------------------------------------------------------------------------
// Aggregation init (vectorized): out[i,f..f+V) = bias[f..] + dinv[i]^2 * h[i,f..)
// ---------------------------------------------------------------------------
template <int VEC>
__global__ void agg_init_vec(const float* __restrict__ h, const float* __restrict__ dinv,
                             const float* __restrict__ bias, float* __restrict__ out,
                             int n, int Nf) {
    const int chunks = Nf / VEC;
    size_t t = (size_t)blockIdx.x * blockDim.x + threadIdx.x;
    if (t >= (size_t)n * chunks) return;
    int i = (int)(t / chunks);
    int f = (int)(t - (size_t)i * chunks) * VEC;
    float di = dinv[i];
    float w  = di * di;
    size_t base = (size_t)i * Nf + f;
    if constexpr (VEC == 4) {
        float4 v = *(const float4*)(h + base);
        float4 o;
        o.x = bias[f + 0] + w * v.x;
        o.y = bias[f + 1] + w * v.y;
        o.z = bias[f + 2] + w * v.z;
        o.w = bias[f + 3] + w * v.w;
        *(float4*)(out + base) = o;
    } else {
        float2 v = *(const float2*)(h + base);
        float2 o;
        o.x = bias[f + 0] + w * v.x;
        o.y = bias[f + 1] + w * v.y;
        *(float2*)(out + base) = o;
    }
}

// ---------------------------------------------------------------------------
// Edge scatter (vectorized): out[dst,f..f+V) += h[src,f..f+V) * dinv[src]*dinv[dst]
// One thread covers VEC features of one edge: one b128 gather of h, one pair of
// int64 index loads, one dinv product, VEC non-returning f32 atomics to one
// destination row (L2-resident reduction).
// ---------------------------------------------------------------------------
template <int VEC>
__global__ void scatter_edges_vec(const float* __restrict__ h, const float* __restrict__ dinv,
                                  const long long* __restrict__ src,
                                  const long long* __restrict__ dst,
                                  float* __restrict__ out, long long E, int Nf) {
    const int chunks = Nf / VEC;
    long long t = (long long)blockIdx.x * blockDim.x + threadIdx.x;
    if (t >= E * (long long)chunks) return;
    long long e = t / chunks;
    int f = (int)(t - e * (long long)chunks) * VEC;
    long long s = src[e];
    long long d = dst[e];
    float w = dinv[s] * dinv[d];
    const float* hp = h + (size_t)s * Nf + f;
    float* op = out + (size_t)d * Nf + f;
    if constexpr (VEC == 4) {
        float4 v = *(const float4*)hp;
        atomicAdd(op + 0, v.x * w);
        atomicAdd(op + 1, v.y * w);
        atomicAdd(op + 2, v.z * w);
        atomicAdd(op + 3, v.w * w);
    } else {
        float2 v = *(const float2*)hp;
        atomicAdd(op + 0, v.x * w);
        atomicAdd(op + 1, v.y * w);
    }
}

// ---------------------------------------------------------------------------
// log_softmax over 2 classes, in place
// ---------------------------------------------------------------------------
__global__ void log_softmax2(float* __restrict__ out, int n) {
    int i = blockIdx.x * blockDim.x + threadIdx.x;
    if (i >= n) return;
    float a = out[2 * i], b = out[2 * i + 1];
    float m = fmaxf(a, b);
    float l = m + logf(expf(a - m) + expf(b - m));
    out[2 * i]     = a - l;
    out[2 * i + 1] = b - l;
}

// ---------------------------------------------------------------------------
extern "C" void kernel_launch(void* const* d_in, const int* in_sizes, int n_in,
                              void* d_out, int out_size, void* d_ws, size_t ws_size,
                              hipStream_t stream) {
    const float* x      = (const float*)d_in[0];
    const long long* ei = (const long long*)d_in[1];   // (2, E) int64 per reference
    const float* W[5] = {(const float*)d_in[2], (const float*)d_in[4],
                         (const float*)d_in[6], (const float*)d_in[8],
                         (const float*)d_in[10]};
    const float* Bz[5] = {(const float*)d_in[3], (const float*)d_in[5],
                          (const float*)d_in[7], (const float*)d_in[9],
                          (const float*)d_in[11]};

    const int n = N_NODES;
    const long long E = (long long)in_sizes[1] / 2;
    const long long* src = ei;
    const long long* dst = ei + E;

    // Workspace layout (floats): dinv[n] | h[n*128] | actA[n*64] | actB[n*128]
    float* ws   = (float*)d_ws;
    float* dinv = ws;
    float* h    = dinv + (size_t)n;
    float* actA = h + (size_t)n * 128;
    float* actB = actA + (size_t)n * 64;
    float* outp = (float*)d_out;

    const int dims[6] = {256, 64, 128, 64, 16, 2};

    // ---- degrees -> dinv (in place) ----
    deg_init<<<(n + 255) / 256, 256, 0, stream>>>(dinv, n);
    deg_accum<<<(unsigned)((E + 255) / 256), 256, 0, stream>>>(dst, dinv, E);
    deg_to_dinv<<<(n + 255) / 256, 256, 0, stream>>>(dinv, n);

    // ---- 5 GCN layers ----
    const float* layer_in = x;
    for (int l = 0; l < 5; ++l) {
        const int K  = dims[l];
        const int Nf = dims[l + 1];

        dim3 grid(n / 16, (Nf + 15) / 16);
        gemm_wmma_f32<<<grid, 32, 0, stream>>>(layer_in, W[l], h, K, Nf, l > 0 ? 1 : 0);

        float* layer_out = (l == 4) ? outp : ((l & 1) ? actB : actA);

        if (Nf % 4 == 0) {
            size_t tot = (size_t)n * (Nf / 4);
            agg_init_vec<4><<<(unsigned)((tot + 255) / 256), 256, 0, stream>>>(
                h, dinv, Bz[l], layer_out, n, Nf);
            long long te = E * (long long)(Nf / 4);
            scatter_edges_vec<4><<<(unsigned)((te + 255) / 256), 256, 0, stream>>>(
                h, dinv, src, dst, layer_out, E, Nf);
        } else {
            size_t tot = (size_t)n * (Nf / 2);
            agg_init_vec<2><<<(unsigned)((tot + 255) / 256), 256, 0, stream>>>(
                h, dinv, Bz[l], layer_out, n, Nf);
            long long te = E * (long long)(Nf / 2);
            scatter_edges_vec<2><<<(unsigned)((te + 255) / 256), 256, 0, stream>>>(
                h, dinv, src, dst, layer_out, E, Nf);
        }
        layer_in = layer_out;
    }

    log_softmax2<<<(n + 255) / 256, 256, 0, stream>>>(outp, n);
}